// TPS_Transformer_67654324847003
// MI455X (gfx1250) — compile-verified
//
#include <hip/hip_runtime.h>
#include <hip/hip_bf16.h>
#include <math.h>

#define KF      0.3f
#define OFF     0.35f      // (1 - 0.3)/2

typedef __attribute__((ext_vector_type(2))) float v2f;
typedef __attribute__((ext_vector_type(8))) float v8f;

// ---------------------------------------------------------------------------
// Kernel 1: per-sample min/max of polargrid (N blocks, 256 threads)
// vmm layout per sample: [minx, miny, maxx, maxy]
// ---------------------------------------------------------------------------
__global__ void tps_minmax_kernel(const float* __restrict__ pg, float* __restrict__ vmm) {
    const int n = blockIdx.x;
    const float* p = pg + (size_t)n * 2048;   // 1024 pixels * 2
    const int t = threadIdx.x;
    float mnx = 1e30f, mny = 1e30f, mxx = -1e30f, mxy = -1e30f;
    for (int i = t; i < 1024; i += 256) {
        v2f xy = *(const v2f*)(p + i * 2);
        mnx = fminf(mnx, xy.x); mxx = fmaxf(mxx, xy.x);
        mny = fminf(mny, xy.y); mxy = fmaxf(mxy, xy.y);
    }
    __shared__ float s0[256], s1[256], s2[256], s3[256];
    s0[t] = mnx; s1[t] = mny; s2[t] = mxx; s3[t] = mxy;
    __syncthreads();
    for (int off = 128; off > 0; off >>= 1) {
        if (t < off) {
            s0[t] = fminf(s0[t], s0[t + off]);
            s1[t] = fminf(s1[t], s1[t + off]);
            s2[t] = fmaxf(s2[t], s2[t + off]);
            s3[t] = fmaxf(s3[t], s3[t + off]);
        }
        __syncthreads();
    }
    if (t == 0) {
        vmm[n * 4 + 0] = s0[0];
        vmm[n * 4 + 1] = s1[0];
        vmm[n * 4 + 2] = s2[0];
        vmm[n * 4 + 3] = s3[0];
    }
}

// ---------------------------------------------------------------------------
// Kernel 2: fp32 GEMM with WMMA (Y = relu(X @ W + b)).
// One wave per 64x16 output tile: 4 M-subtiles share one B tile, so the
// strided B loads are amortized 4x. K stepped by 4 via V_WMMA_F32_16X16X4_F32.
// A lane layout: m = lane%16, k = vgpr + 2*(lane/16)  -> contiguous float2
// B lane layout: n = lane%16, k = vgpr + 2*(lane/16)
// D: VGPR r holds row r + 8*(lane/16), column lane%16.
// ---------------------------------------------------------------------------
__global__ void tps_gemm_bias_relu_wmma(const float* __restrict__ X,
                                        const float* __restrict__ Wt,
                                        const float* __restrict__ bias,
                                        float* __restrict__ Y,
                                        int M, int K, int N) {
    const int wave = (int)((blockIdx.x * blockDim.x + threadIdx.x) >> 5);
    const int lane = threadIdx.x & 31;
    const int tiles_n  = N >> 4;
    const int tiles_m4 = M >> 6;                  // 64-row tiles
    if (wave >= tiles_m4 * tiles_n) return;       // wave-uniform guard

    const int tm  = wave / tiles_n;               // 64-row tile index
    const int tn  = wave % tiles_n;
    const int l16 = lane & 15;
    const int hi  = lane >> 4;                    // 0 or 1
    const int koff = hi * 2;

    const int bcol = tn * 16 + l16;

    const float* xr0 = X + (size_t)(tm * 64 +      l16) * K + koff;
    const float* xr1 = X + (size_t)(tm * 64 + 16 + l16) * K + koff;
    const float* xr2 = X + (size_t)(tm * 64 + 32 + l16) * K + koff;
    const float* xr3 = X + (size_t)(tm * 64 + 48 + l16) * K + koff;
    const float* wcol = Wt + (size_t)koff * N + bcol;

    v8f acc0 = {}, acc1 = {}, acc2 = {}, acc3 = {};
    for (int k = 0; k < K; k += 4) {
        v2f b;
        b.x = wcol[(size_t)k * N];                // W[k+koff  ][bcol]
        b.y = wcol[(size_t)(k + 1) * N];          // W[k+koff+1][bcol]
        v2f a0 = *(const v2f*)(xr0 + k);
        v2f a1 = *(const v2f*)(xr1 + k);
        v2f a2 = *(const v2f*)(xr2 + k);
        v2f a3 = *(const v2f*)(xr3 + k);
        acc0 = __builtin_amdgcn_wmma_f32_16x16x4_f32(false, a0, false, b, (short)0, acc0, false, false);
        acc1 = __builtin_amdgcn_wmma_f32_16x16x4_f32(false, a1, false, b, (short)0, acc1, false, false);
        acc2 = __builtin_amdgcn_wmma_f32_16x16x4_f32(false, a2, false, b, (short)0, acc2, false, false);
        acc3 = __builtin_amdgcn_wmma_f32_16x16x4_f32(false, a3, false, b, (short)0, acc3, false, false);
    }

    const float bv = bias[bcol];
    float* y0 = Y + (size_t)(tm * 64 + hi * 8) * N + bcol;
#pragma unroll
    for (int r = 0; r < 8; ++r) {
        float v0 = acc0[r] + bv; v0 = v0 > 0.f ? v0 : 0.f;
        float v1 = acc1[r] + bv; v1 = v1 > 0.f ? v1 : 0.f;
        float v2 = acc2[r] + bv; v2 = v2 > 0.f ? v2 : 0.f;
        float v3 = acc3[r] + bv; v3 = v3 > 0.f ? v3 : 0.f;
        y0[(size_t)(r     ) * N] = v0;
        y0[(size_t)(r + 16) * N] = v1;
        y0[(size_t)(r + 32) * N] = v2;
        y0[(size_t)(r + 48) * N] = v3;
    }
}

// ---------------------------------------------------------------------------
// Kernel 3: final tiny layer (K=256, N=36) + tanh. One thread per output.
// ---------------------------------------------------------------------------
__global__ void tps_final_tanh_kernel(const float* __restrict__ act,
                                      const float* __restrict__ W4,
                                      const float* __restrict__ b4,
                                      float* __restrict__ theta,
                                      int M, int K, int N) {
    const int idx = blockIdx.x * blockDim.x + threadIdx.x;
    if (idx >= M * N) return;
    const int m = idx / N, n = idx % N;
    const float* x = act + (size_t)m * K;
    float s = b4[n];
    for (int k = 0; k < K; ++k) s = fmaf(x[k], W4[(size_t)k * N + n], s);
    theta[idx] = tanhf(s);
}

// ---------------------------------------------------------------------------
// Kernel 4: TPS warp field. One block per sample; 256 threads over 1024 px.
// ---------------------------------------------------------------------------
__global__ void tps_warp_field_kernel(const float* __restrict__ pgAll,
                                      const float* __restrict__ vmm,
                                      const float* __restrict__ theta,
                                      float* __restrict__ tw) {
    const int n = blockIdx.x;
    const float* pg = pgAll + (size_t)n * 2048;
    const int t = threadIdx.x;

    __shared__ float ctrlx[16], ctrly[16];
    __shared__ float wsx[16], wsy[16];
    __shared__ float ax[3], ay[3];

    const float vminx = vmm[n * 4 + 0];
    const float vminy = vmm[n * 4 + 1];
    const float ptpx  = vmm[n * 4 + 2] - vminx;
    const float ptpy  = vmm[n * 4 + 3] - vminy;

    if (t < 16) {                                  // ctrl points: rows/cols 0,8,16,24
        const int hh = (t >> 2) * 8, ww = (t & 3) * 8;
        v2f xy = *(const v2f*)(pg + (hh * 32 + ww) * 2);
        ctrlx[t] = (xy.x - vminx) / ptpx * KF + OFF;
        ctrly[t] = (xy.y - vminy) / ptpy * KF + OFF;
    }
    if (t == 32 || t == 33) {                      // build w (16x2) and a (3x2)
        const int c = t - 32;
        const float* th = theta + (size_t)n * 36;
        float sum = 0.f;
        for (int p = 0; p < 15; ++p) sum += th[p * 2 + c];
        float* wv = (c == 0) ? wsx : wsy;
        wv[0] = -sum;
        for (int p = 1; p < 16; ++p) wv[p] = th[(p - 1) * 2 + c];
        float* av = (c == 0) ? ax : ay;
        av[0] = th[15 * 2 + c];
        av[1] = th[16 * 2 + c];
        av[2] = th[17 * 2 + c];
    }
    __syncthreads();

    for (int p = t; p < 1024; p += 256) {
        v2f pxy = *(const v2f*)(pg + p * 2);
        const float pgx = (pxy.x - vminx) / ptpx * KF + OFF;
        const float pgy = (pxy.y - vminy) / ptpy * KF + OFF;
        float bx = 0.f, by = 0.f;
#pragma unroll
        for (int k = 0; k < 16; ++k) {
            const float dx = pgx - ctrlx[k];
            const float dy = pgy - ctrly[k];
            const float d2 = dx * dx + dy * dy;
            const float U  = d2 * logf(sqrtf(d2) + 1e-6f);
            bx = fmaf(U, wsx[k], bx);
            by = fmaf(U, wsy[k], by);
        }
        const float zx = ax[0] + pgx * ax[1] + pgy * ax[2] + bx;
        const float zy = ay[0] + pgx * ay[1] + pgy * ay[2] + by;
        v2f twv;
        twv.x = ((pgx + zx) - OFF) / KF * ptpx + vminx;
        twv.y = ((pgy + zy) - OFF) / KF * ptpy + vminy;
        *(v2f*)(tw + ((size_t)n * 1024 + p) * 2) = twv;
    }
}

// ---------------------------------------------------------------------------
// Kernel 5: bilinear grid-sample (border clamp). One thread per pixel,
// loops all 128 channels: weights computed once, writes fully coalesced.
// imgs (16.7 MB) and tw (8 MB) are L2-resident; bound by 536 MB output.
// ---------------------------------------------------------------------------
__global__ void tps_grid_sample_kernel(const float* __restrict__ imgs,
                                       const int* __restrict__ imgIDs,
                                       const float* __restrict__ tw,
                                       float* __restrict__ out) {
    const int pix = blockIdx.x * blockDim.x + threadIdx.x;   // 0 .. N*H*W-1
    if (pix >= 1024 * 1024) return;
    const int n  = pix >> 10;
    const int hw = pix & 1023;
    const int C = 128, Hi = 64, Wi = 64;

    v2f twv = *(const v2f*)(tw + (size_t)pix * 2);

    float ix = ((twv.x + 1.f) * (float)Wi - 1.f) * 0.5f;
    float iy = ((twv.y + 1.f) * (float)Hi - 1.f) * 0.5f;
    ix = fminf(fmaxf(ix, 0.f), (float)(Wi - 1));
    iy = fminf(fmaxf(iy, 0.f), (float)(Hi - 1));

    const float x0f = floorf(ix), y0f = floorf(iy);
    const float wx = ix - x0f,    wy = iy - y0f;
    const int x0 = (int)x0f,      y0 = (int)y0f;
    const int x1 = min(x0 + 1, Wi - 1);
    const int y1 = min(y0 + 1, Hi - 1);

    const int o00 = y0 * Wi + x0, o01 = y0 * Wi + x1;
    const int o10 = y1 * Wi + x0, o11 = y1 * Wi + x1;
    const float w00 = (1.f - wx) * (1.f - wy), w01 = wx * (1.f - wy);
    const float w10 = (1.f - wx) * wy,         w11 = wx * wy;

    const float* img = imgs + (size_t)imgIDs[n] * C * Hi * Wi;
    float* op = out + (size_t)n * C * 1024 + hw;

#pragma unroll 4
    for (int c = 0; c < C; ++c) {
        const float* ic = img + (size_t)c * (Hi * Wi);
        float v = ic[o00] * w00 + ic[o01] * w01 + ic[o10] * w10 + ic[o11] * w11;
        op[(size_t)c * 1024] = v;
    }
}

// ---------------------------------------------------------------------------
extern "C" void kernel_launch(void* const* d_in, const int* in_sizes, int n_in,
                              void* d_out, int out_size, void* d_ws, size_t ws_size,
                              hipStream_t stream) {
    const float* polargrid = (const float*)d_in[0];   // (1024,32,32,2)
    const float* xt        = (const float*)d_in[1];   // (1024,128,5,5) -> (1024,3200)
    const float* imgs      = (const float*)d_in[2];   // (8,128,64,64)
    const int*   imgIDs    = (const int*)  d_in[3];   // (1024,)
    const float* W1        = (const float*)d_in[4];   // (3200,512)
    const float* b1        = (const float*)d_in[5];
    const float* W2        = (const float*)d_in[6];   // (512,256)
    const float* b2        = (const float*)d_in[7];
    const float* W3        = (const float*)d_in[8];   // (256,256)
    const float* b3        = (const float*)d_in[9];
    const float* W4        = (const float*)d_in[10];  // (256,36)
    const float* b4        = (const float*)d_in[11];
    float* out = (float*)d_out;                       // (1024,128,32,32)

    const int N = 1024;

    // workspace carve-up (floats)
    float* wsf   = (float*)d_ws;
    float* vmm   = wsf;                       // 1024*4        =    4096
    float* act1  = vmm  + 4096;               // 1024*512      =  524288
    float* act2  = act1 + 524288;             // 1024*256      =  262144
    float* act3  = act2 + 262144;             // 1024*256      =  262144
    float* theta = act3 + 262144;             // 1024*36       =   36864
    float* tw    = theta + 36864;             // 1024*1024*2   = 2097152

    // 1) per-sample min/max
    tps_minmax_kernel<<<N, 256, 0, stream>>>(polargrid, vmm);

    // 2) MLP via fp32 WMMA (one wave per 64x16 tile; 8 waves/block)
    {   // L1: 1024x3200 @ 3200x512  -> 16*32 = 512 waves -> 64 blocks
        tps_gemm_bias_relu_wmma<<<64, 256, 0, stream>>>(xt,   W1, b1, act1, 1024, 3200, 512);
        // L2: 1024x512 @ 512x256    -> 16*16 = 256 waves -> 32 blocks
        tps_gemm_bias_relu_wmma<<<32, 256, 0, stream>>>(act1, W2, b2, act2, 1024, 512, 256);
        // L3: 1024x256 @ 256x256
        tps_gemm_bias_relu_wmma<<<32, 256, 0, stream>>>(act2, W3, b3, act3, 1024, 256, 256);
    }

    // 3) final layer (N=36, not 16-aligned) + tanh
    tps_final_tanh_kernel<<<(1024 * 36 + 255) / 256, 256, 0, stream>>>(
        act3, W4, b4, theta, 1024, 256, 36);

    // 4) TPS warp field (one block per sample)
    tps_warp_field_kernel<<<N, 256, 0, stream>>>(polargrid, vmm, theta, tw);

    // 5) bilinear grid-sample: 1M pixel-threads x 128 channels
    tps_grid_sample_kernel<<<(1024 * 1024) / 256, 256, 0, stream>>>(imgs, imgIDs, tw, out);
}